// GNNA_GIN_50697793962358
// MI455X (gfx1250) — compile-verified
//
#include <hip/hip_runtime.h>
#include <hip/hip_bf16.h>
#include <math.h>

typedef __attribute__((ext_vector_type(2))) float v2f;
typedef __attribute__((ext_vector_type(8))) float v8f;

#define DIN 128
#define DH  128
#define DOUT 40
#define XS_LD 132   // padded LDS row stride (floats): kills 16-way bank conflicts, keeps 16B alignment

static inline int cdiv_i(long long a, int b) { return (int)((a + b - 1) / b); }

// ---------------- degree / normalization ----------------

__global__ void k_fill1(float* __restrict__ p, int n) {
  int i = blockIdx.x * blockDim.x + threadIdx.x;
  if (i < n) p[i] = 1.0f;
}

__global__ void k_deg_scatter(float* __restrict__ deg,
                              const long long* __restrict__ dst, int E) {
  int e = blockIdx.x * blockDim.x + threadIdx.x;
  if (e < E) atomicAdd(&deg[(int)dst[e]], 1.0f);
}

__global__ void k_rsqrt(float* __restrict__ norm, const float* __restrict__ deg, int n) {
  int i = blockIdx.x * blockDim.x + threadIdx.x;
  if (i < n) norm[i] = rsqrtf(deg[i]);
}

// ---------------- GEMM1: H[n,128] = X[n,128] @ W[128,128] via fp32 WMMA ----------------
// Block = 256 threads (8 waves). Block b owns rows [16b,16b+16); wave w owns cols [16w,16w+16).
// A (16x4 f32): lane<16 -> row=lane, K pair {k,k+1}; lane>=16 -> row=lane-16, K pair {k+2,k+3}.
// B (4x16 f32): VGPR j, lanes0-15 -> K row j; lanes16-31 -> K row j+2.
// C (16x16 f32): VGPR j, lanes0-15 -> M=j, N=lane; lanes16-31 -> M=j+8, N=lane-16.

__global__ __launch_bounds__(256) void k_gemm1(const float* __restrict__ X,
                                               const float* __restrict__ W,
                                               float* __restrict__ H) {
  __shared__ float xs[16 * XS_LD];
  const int tid  = threadIdx.x;
  const int row0 = blockIdx.x * 16;

  // stage 16x128 X tile as float4 (2 b128 loads/thread), coalesced
  {
    const float4* Xv = (const float4*)(X + (size_t)row0 * DIN);
    #pragma unroll
    for (int f = 0; f < 2; ++f) {
      int fidx = tid + f * 256;            // 0..511 float4s
      int r = fidx >> 5;                   // /32 float4s per row
      int c = (fidx & 31) * 4;
      *(float4*)&xs[r * XS_LD + c] = Xv[fidx];
    }
  }
  __syncthreads();

  const int wave = tid >> 5;
  const int lane = tid & 31;
  const int col0 = wave * 16;
  const int half = lane >> 4;   // 0: K{+0,+1}, 1: K{+2,+3}
  const int l15  = lane & 15;

  v8f c = {};
  #pragma unroll
  for (int k = 0; k < DIN; k += 4) {
    const int kk = k + half * 2;
    v2f a;
    a.x = xs[l15 * XS_LD + kk];
    a.y = xs[l15 * XS_LD + kk + 1];
    v2f b;
    b.x = W[kk * DH + col0 + l15];
    b.y = W[(kk + 1) * DH + col0 + l15];
    c = __builtin_amdgcn_wmma_f32_16x16x4_f32(false, a, false, b, (short)0, c,
                                              false, false);
  }

  #pragma unroll
  for (int j = 0; j < 8; ++j) {
    const int r = row0 + j + half * 8;
    H[r * DH + col0 + l15] = c[j];
  }
}

// ---------------- layer-1 aggregation ----------------

__global__ void k_self128(float* __restrict__ agg, const float* __restrict__ h,
                          const float* __restrict__ norm, int n) {
  int gid = blockIdx.x * blockDim.x + threadIdx.x;
  if (gid >= n * DH) return;
  int i = gid >> 7;
  float nv = norm[i];
  agg[gid] = h[gid] * nv * nv;
}

// one wave32 per edge: lane handles 4 consecutive channels -> coalesced 512B row segments
__global__ void k_scatter128(float* __restrict__ agg, const float* __restrict__ h,
                             const float* __restrict__ norm,
                             const long long* __restrict__ src,
                             const long long* __restrict__ dst, int E) {
  int gid = blockIdx.x * blockDim.x + threadIdx.x;
  int e = gid >> 5;
  if (e >= E) return;
  int lane = gid & 31;
  int s = (int)src[e];
  int d = (int)dst[e];
  float coef = norm[s] * norm[d];
  const float4 hv = *(const float4*)(h + (size_t)s * DH + lane * 4);
  float* ap = agg + (size_t)d * DH + lane * 4;
  atomicAdd(ap + 0, hv.x * coef);
  atomicAdd(ap + 1, hv.y * coef);
  atomicAdd(ap + 2, hv.z * coef);
  atomicAdd(ap + 3, hv.w * coef);
}

__global__ void k_bias_relu(float* __restrict__ agg, const float* __restrict__ b, int n) {
  int gid = blockIdx.x * blockDim.x + threadIdx.x;
  if (gid >= n * DH) return;
  float v = agg[gid] + b[gid & 127];
  agg[gid] = v > 0.0f ? v : 0.0f;
}

// ---------------- GEMM2: Out[n,40] = Hact[n,128] @ W2[128,40] via fp32 WMMA ----------------
// Block = 96 threads (3 waves, one per 16-column tile; tile 2 has 8 valid columns).

__global__ __launch_bounds__(96) void k_gemm2(const float* __restrict__ Hin,
                                              const float* __restrict__ W,
                                              float* __restrict__ Out) {
  __shared__ float xs[16 * XS_LD];
  const int tid  = threadIdx.x;
  const int row0 = blockIdx.x * 16;

  // stage 16x128 activation tile as float4 (512 float4s across 96 threads)
  {
    const float4* Hv = (const float4*)(Hin + (size_t)row0 * DH);
    for (int fidx = tid; fidx < 512; fidx += 96) {
      int r = fidx >> 5;
      int c = (fidx & 31) * 4;
      *(float4*)&xs[r * XS_LD + c] = Hv[fidx];
    }
  }
  __syncthreads();

  const int wave = tid >> 5;
  const int lane = tid & 31;
  const int col0 = wave * 16;
  const int half = lane >> 4;
  const int l15  = lane & 15;
  const int col  = col0 + l15;
  const bool ok  = (col < DOUT);

  v8f c = {};
  #pragma unroll
  for (int k = 0; k < DH; k += 4) {
    const int kk = k + half * 2;
    v2f a;
    a.x = xs[l15 * XS_LD + kk];
    a.y = xs[l15 * XS_LD + kk + 1];
    v2f b;
    b.x = ok ? W[kk * DOUT + col] : 0.0f;
    b.y = ok ? W[(kk + 1) * DOUT + col] : 0.0f;
    c = __builtin_amdgcn_wmma_f32_16x16x4_f32(false, a, false, b, (short)0, c,
                                              false, false);
  }

  if (ok) {
    #pragma unroll
    for (int j = 0; j < 8; ++j) {
      const int r = row0 + j + half * 8;
      Out[r * DOUT + col] = c[j];
    }
  }
}

// ---------------- layer-2 aggregation + log-softmax ----------------

__global__ void k_self40(float* __restrict__ agg, const float* __restrict__ o,
                         const float* __restrict__ norm, int n) {
  int gid = blockIdx.x * blockDim.x + threadIdx.x;
  if (gid >= n * DOUT) return;
  int i = gid / DOUT;
  float nv = norm[i];
  agg[gid] = o[gid] * nv * nv;
}

__global__ void k_scatter40(float* __restrict__ agg, const float* __restrict__ o,
                            const float* __restrict__ norm,
                            const long long* __restrict__ src,
                            const long long* __restrict__ dst, int E) {
  int gid = blockIdx.x * blockDim.x + threadIdx.x;
  if (gid >= E * DOUT) return;
  int e = gid / DOUT;
  int c = gid - e * DOUT;
  int s = (int)src[e];
  int d = (int)dst[e];
  float coef = norm[s] * norm[d];
  atomicAdd(agg + (size_t)d * DOUT + c, o[(size_t)s * DOUT + c] * coef);
}

__global__ void k_logsoftmax(float* __restrict__ out, const float* __restrict__ agg,
                             const float* __restrict__ b, int n) {
  int i = blockIdx.x * blockDim.x + threadIdx.x;
  if (i >= n) return;
  float v[DOUT];
  float m = -INFINITY;
  #pragma unroll
  for (int c = 0; c < DOUT; ++c) {
    v[c] = agg[(size_t)i * DOUT + c] + b[c];
    m = fmaxf(m, v[c]);
  }
  float s = 0.0f;
  #pragma unroll
  for (int c = 0; c < DOUT; ++c) s += expf(v[c] - m);
  float ls = logf(s);
  #pragma unroll
  for (int c = 0; c < DOUT; ++c) out[(size_t)i * DOUT + c] = v[c] - m - ls;
}

// ---------------- launcher ----------------

extern "C" void kernel_launch(void* const* d_in, const int* in_sizes, int n_in,
                              void* d_out, int out_size, void* d_ws, size_t ws_size,
                              hipStream_t stream) {
  const float*     X  = (const float*)d_in[0];
  const float*     W1 = (const float*)d_in[1];
  const float*     b1 = (const float*)d_in[2];
  const float*     W2 = (const float*)d_in[3];
  const float*     b2 = (const float*)d_in[4];
  const long long* EI = (const long long*)d_in[5];

  const int n = in_sizes[0] / DIN;       // 50000 (divisible by 16)
  const int E = in_sizes[5] / 2;         // 800000
  const long long* src = EI;
  const long long* dst = EI + E;

  char*  ws  = (char*)d_ws;
  size_t off = 0;
  auto carve = [&](size_t bytes) -> float* {
    float* p = (float*)(ws + off);
    off += (bytes + 255) & ~(size_t)255;
    return p;
  };
  float* h1   = carve((size_t)n * DH   * sizeof(float));  // X@W1
  float* agg1 = carve((size_t)n * DH   * sizeof(float));  // layer1 agg / relu out
  float* ol   = carve((size_t)n * DOUT * sizeof(float));  // Hact@W2
  float* agg2 = carve((size_t)n * DOUT * sizeof(float));  // layer2 agg
  float* deg  = carve((size_t)n * sizeof(float));
  float* nrm  = carve((size_t)n * sizeof(float));
  (void)ws_size;

  float* out = (float*)d_out;

  // degrees and symmetric normalization
  k_fill1<<<cdiv_i(n, 256), 256, 0, stream>>>(deg, n);
  k_deg_scatter<<<cdiv_i(E, 256), 256, 0, stream>>>(deg, dst, E);
  k_rsqrt<<<cdiv_i(n, 256), 256, 0, stream>>>(nrm, deg, n);

  // layer 1: dense GEMM (WMMA fp32), self-loop, edge scatter, bias+relu
  k_gemm1<<<n / 16, 256, 0, stream>>>(X, W1, h1);
  k_self128<<<cdiv_i((long long)n * DH, 256), 256, 0, stream>>>(agg1, h1, nrm, n);
  k_scatter128<<<cdiv_i((long long)E * 32, 256), 256, 0, stream>>>(agg1, h1, nrm, src, dst, E);
  k_bias_relu<<<cdiv_i((long long)n * DH, 256), 256, 0, stream>>>(agg1, b1, n);

  // layer 2: dense GEMM (WMMA fp32), self-loop, edge scatter, bias + log_softmax
  k_gemm2<<<n / 16, 96, 0, stream>>>(agg1, W2, ol);
  k_self40<<<cdiv_i((long long)n * DOUT, 256), 256, 0, stream>>>(agg2, ol, nrm, n);
  k_scatter40<<<cdiv_i((long long)E * DOUT, 256), 256, 0, stream>>>(agg2, ol, nrm, src, dst, E);
  k_logsoftmax<<<cdiv_i(n, 256), 256, 0, stream>>>(out, agg2, b2, n);
}